// LinearAttentionHead_79903571574891
// MI455X (gfx1250) — compile-verified
//
#include <hip/hip_runtime.h>
#include <math.h>

typedef __attribute__((ext_vector_type(2))) float v2f;
typedef __attribute__((ext_vector_type(4))) float v4f;
typedef __attribute__((ext_vector_type(8))) float v8f;

#define BATCH 16
#define SEQ   2048
#define DIM   64
#define SCALE 0.125f          // 1/sqrt(64)
#define ROWSTRIDE 2052        // padded LDS row stride (16B-multiple, shifts banks 4/row)
#define NT (SEQ / 16)         // 128 column tiles of 16 keys
#define NW 4                  // waves per block (wave32)

__global__ __launch_bounds__(128)
void attn_fp32_wmma_kernel(const float* __restrict__ Q,
                           const float* __restrict__ K,
                           const float* __restrict__ V,
                           float* __restrict__ Out,
                           float* __restrict__ Attn) {
  // 16 x 2048 score panel for this Q row-block lives entirely in LDS (CDNA5: 320KB/WGP)
  __shared__ alignas(16) float sP[16 * ROWSTRIDE];   // ~131 KB
  __shared__ float sO[16 * DIM];                     // 4 KB output accumulator
  __shared__ float sInv[16];                         // 1/rowsum

  const int tid   = threadIdx.x;
  const int lane  = tid & 31;
  const int w     = tid >> 5;
  const int b     = blockIdx.x >> 7;     // batch index (128 row-blocks per batch)
  const int rb    = blockIdx.x & 127;    // row-block within batch
  const int qbase = rb * 16;

  const int half = (lane >> 4) & 1;      // 0: lanes 0-15, 1: lanes 16-31
  const int l15  = lane & 15;
  const int koff = half * 2;             // A/B operands: upper half-wave carries K+2,K+3

  // ---------------- Pass A: S = (scale*Q) . K^T  via v_wmma_f32_16x16x4_f32 ----
  // A operand (Q tile, 16x4 per step): lane holds row M=l15, features 4s+koff, +1.
  // Fold 1/sqrt(d) into Q once instead of scaling every score tile.
  v2f qa[16];
  {
    const float* qrow = Q + (size_t)(b * SEQ + qbase + l15) * DIM;
#pragma unroll
    for (int s = 0; s < 16; ++s) {
      v2f q = *reinterpret_cast<const v2f*>(qrow + 4 * s + koff);
      qa[s].x = q.x * SCALE;
      qa[s].y = q.y * SCALE;
    }
  }

  for (int ct = w; ct < NT; ct += NW) {
    // B operand (K^T tile, 4x16 per step): lane holds column n=l15 -> key ct*16+n
    v2f kb[16];
    const float* krow = K + (size_t)(b * SEQ + ct * 16 + l15) * DIM;
#pragma unroll
    for (int s = 0; s < 16; ++s)
      kb[s] = *reinterpret_cast<const v2f*>(krow + 4 * s + koff);

    // two independent accumulator chains (even/odd K-steps) to overlap WMMAs
    v8f c0 = {0.f, 0.f, 0.f, 0.f, 0.f, 0.f, 0.f, 0.f};
    v8f c1 = {0.f, 0.f, 0.f, 0.f, 0.f, 0.f, 0.f, 0.f};
#pragma unroll
    for (int s = 0; s < 8; ++s) {
      c0 = __builtin_amdgcn_wmma_f32_16x16x4_f32(false, qa[2 * s + 0], false, kb[2 * s + 0],
                                                 (short)0, c0, false, false);
      c1 = __builtin_amdgcn_wmma_f32_16x16x4_f32(false, qa[2 * s + 1], false, kb[2 * s + 1],
                                                 (short)0, c1, false, false);
    }

    // C layout: VGPR r -> row r (lanes 0-15) / row r+8 (lanes 16-31), col = l15
    const int colbase = ct * 16 + l15;
#pragma unroll
    for (int r = 0; r < 8; ++r) {
      int row = r + half * 8;
      sP[row * ROWSTRIDE + colbase] = c0[r] + c1[r];
    }
  }

  // zero the cross-wave output accumulator while we're pre-barrier anyway
  for (int i = tid; i < 16 * DIM; i += 128) sO[i] = 0.0f;
  __syncthreads();

  // ---------------- Pass B1: row-wise softmax over the LDS panel (b128 LDS ops) --
  // wave w owns rows 4w .. 4w+3; wave32 shuffle reductions
  for (int r = w * 4; r < w * 4 + 4; ++r) {
    float* prow = sP + r * ROWSTRIDE;

    float mx = -3.402823466e38f;
    for (int cidx = lane * 4; cidx < SEQ; cidx += 128) {
      v4f v = *reinterpret_cast<const v4f*>(prow + cidx);
      mx = fmaxf(mx, fmaxf(fmaxf(v.x, v.y), fmaxf(v.z, v.w)));
    }
#pragma unroll
    for (int o = 16; o > 0; o >>= 1) mx = fmaxf(mx, __shfl_xor(mx, o, 32));

    float sum = 0.0f;
    for (int cidx = lane * 4; cidx < SEQ; cidx += 128) {
      v4f v = *reinterpret_cast<const v4f*>(prow + cidx);
      v.x = __expf(v.x - mx);
      v.y = __expf(v.y - mx);
      v.z = __expf(v.z - mx);
      v.w = __expf(v.w - mx);
      *reinterpret_cast<v4f*>(prow + cidx) = v;
      sum += (v.x + v.y) + (v.z + v.w);
    }
#pragma unroll
    for (int o = 16; o > 0; o >>= 1) sum += __shfl_xor(sum, o, 32);
    if (lane == 0) sInv[r] = 1.0f / sum;
  }
  __syncthreads();

  // ---------------- Write normalized attention weights (the HBM-bound part) -----
  {
    float* abase = Attn + (size_t)b * SEQ * SEQ + (size_t)qbase * SEQ;
    for (int r = w * 4; r < w * 4 + 4; ++r) {
      const float inv = sInv[r];
      const float* prow = sP + r * ROWSTRIDE;
      float* arow = abase + (size_t)r * SEQ;
      for (int cidx = lane * 4; cidx < SEQ; cidx += 128) {
        v4f v = *reinterpret_cast<const v4f*>(prow + cidx);   // ds_load_b128
        v.x *= inv; v.y *= inv; v.z *= inv; v.w *= inv;
        *reinterpret_cast<v4f*>(arow + cidx) = v;             // coalesced b128 store
      }
    }
  }

  // ---------------- Pass B2: Out = P_norm . V  (4 interleaved WMMA chains) ------
  v8f acc[4];
#pragma unroll
  for (int t = 0; t < 4; ++t)
    acc[t] = (v8f){0.f, 0.f, 0.f, 0.f, 0.f, 0.f, 0.f, 0.f};

  const float invl = sInv[l15];                   // row normalization for A rows
  const float* vbase = V + (size_t)(b * SEQ) * DIM;

  for (int ct = w; ct < NT; ct += NW) {
    const int keybase = ct * 16;

    // A operand (P tile): row M=l15, step s covers keys 4s+koff, +1 (from LDS)
    v2f pa[4];
#pragma unroll
    for (int s = 0; s < 4; ++s) {
      v2f p = *reinterpret_cast<const v2f*>(sP + l15 * ROWSTRIDE + keybase + 4 * s + koff);
      pa[s].x = p.x * invl;
      pa[s].y = p.y * invl;
    }

    // Preload all V B-operands for this key tile: vb[t][s] = V[key..key+1][16t + l15]
    v2f vb[4][4];
#pragma unroll
    for (int t = 0; t < 4; ++t) {
#pragma unroll
      for (int s = 0; s < 4; ++s) {
        int key = keybase + 4 * s + koff;
        vb[t][s].x = vbase[(size_t)key * DIM + t * 16 + l15];
        vb[t][s].y = vbase[(size_t)(key + 1) * DIM + t * 16 + l15];
      }
    }

    // s outer / t inner: consecutive WMMAs hit independent accumulators
#pragma unroll
    for (int s = 0; s < 4; ++s) {
#pragma unroll
      for (int t = 0; t < 4; ++t)
        acc[t] = __builtin_amdgcn_wmma_f32_16x16x4_f32(false, pa[s], false, vb[t][s],
                                                       (short)0, acc[t], false, false);
    }
  }

  // reduce the 4 waves' partial outputs via LDS float atomics (ds_add_f32)
#pragma unroll
  for (int t = 0; t < 4; ++t) {
#pragma unroll
    for (int r = 0; r < 8; ++r) {
      int row = r + half * 8;
      atomicAdd(&sO[row * DIM + t * 16 + l15], acc[t][r]);
    }
  }
  __syncthreads();

  // coalesced final store of the 16x64 output block
  {
    float* obase = Out + (size_t)(b * SEQ + qbase) * DIM;
    for (int i = tid; i < 16 * DIM; i += 128) obase[i] = sO[i];
  }
}

extern "C" void kernel_launch(void* const* d_in, const int* in_sizes, int n_in,
                              void* d_out, int out_size, void* d_ws, size_t ws_size,
                              hipStream_t stream) {
  const float* Q = (const float*)d_in[0];
  const float* K = (const float*)d_in[1];
  const float* V = (const float*)d_in[2];

  float* Out  = (float*)d_out;                                   // [16,2048,64]
  float* Attn = (float*)d_out + (size_t)BATCH * SEQ * DIM;       // [16,2048,2048]

  dim3 grid(BATCH * (SEQ / 16));   // 2048 workgroups: one per (batch, 16-row block)
  dim3 block(128);                 // 4 wave32 waves
  hipLaunchKernelGGL(attn_fp32_wmma_kernel, grid, block, 0, stream,
                     Q, K, V, Out, Attn);
}